// SH_Msg_37606733644280
// MI455X (gfx1250) — compile-verified
//
#include <hip/hip_runtime.h>

typedef __attribute__((ext_vector_type(2))) float v2f;
typedef __attribute__((ext_vector_type(4))) float v4f;
typedef __attribute__((ext_vector_type(8))) float v8f;

#define SH_DIM 64
#define OUT_L  8

// One wave32 processes a tile of 16 edges.
//   D[l, edge] = sum_k P[k,l] * temp[edge,k]   via 16 chained
//   v_wmma_f32_16x16x4_f32 with A = projection (M=l), B = products (N=edge).
//
//   Load phase: lane L handles edge (L%16); lanes L and L+16 split each
//     64-float row into k%4 in {0,1} / {2,3} halves (contiguous float2 loads,
//     zero redundancy, coalesced per edge row).
//   A-fragment: P[k,m] = (m*m <= k < (m+1)*(m+1)) synthesized from a per-lane
//     64-bit mask pre-shifted by 2*half -> per chunk just bfe+cvt at
//     compile-time bit offsets.
//   Store phase: lane L (<16) holds out[edge, 0..7] in acc[0..7] ->
//     two coalesced global_store_b128 per wave.
__global__ __launch_bounds__(256) void sh_irrep_inner_kernel(
    const long long* __restrict__ edge_index,  // [2, E] int64
    const float*     __restrict__ node_sh,     // [N_NODES, 64] f32
    float*           __restrict__ out,         // [E, 8] f32
    int E)
{
    const int lane = threadIdx.x & 31;
    const int wave = threadIdx.x >> 5;
    const long long tile = (long long)blockIdx.x * 8 + wave;
    const long long base = tile * 16;
    if (base >= E) return;                 // uniform per wave: EXEC stays full

    const int eLocal = lane & 15;          // edge within tile (B column N)
    const int half   = lane >> 4;          // which k%4 pair this lane covers
    const int m      = eLocal;             // A-matrix row M this lane supplies

    const long long e0 = base + eLocal;
    long long e = e0 < E ? e0 : (long long)(E - 1);  // clamp loads (partial tile)
    const int row = (int)edge_index[e];
    const int col = (int)edge_index[(long long)E + e];

    const float* __restrict__ arow = node_sh + (long long)row * SH_DIM;
    const float* __restrict__ brow = node_sh + (long long)col * SH_DIM;

    // Per-lane projection bitmask over k=0..63: bit k set iff m^2 <= k < (m+1)^2.
    unsigned long long mask = 0ull;
    if (m < OUT_L) {
        const int klo = m * m;              // <= 49
        const int khi = (m + 1) * (m + 1);  // <= 64
        const unsigned long long top =
            (khi >= 64) ? ~0ull : ((1ull << khi) - 1ull);
        mask = top & ~((1ull << klo) - 1ull);
    }
    mask >>= (2 * half);                    // fold in this lane's k%4 pair
    const unsigned mlo = (unsigned)mask;
    const unsigned mhi = (unsigned)(mask >> 32);

    v8f acc = {0.f, 0.f, 0.f, 0.f, 0.f, 0.f, 0.f, 0.f};

#pragma unroll
    for (int c = 0; c < 16; ++c) {
        const int k0 = 4 * c + 2 * half;    // global K of vector component 0

        // B-matrix fragment: elementwise product (contiguous float2 per lane)
        v2f b;
        b.x = arow[k0]     * brow[k0];
        b.y = arow[k0 + 1] * brow[k0 + 1];

        // A-matrix fragment: projection bits at compile-time offsets -> bfe+cvt
        const unsigned src = (c < 8) ? mlo : mhi;
        const int      sh  = (c < 8) ? (4 * c) : (4 * c - 32);
        v2f a;
        a.x = (float)((src >> sh) & 1u);
        a.y = (float)((src >> (sh + 1)) & 1u);

        // 8 args: (neg_a, A, neg_b, B, c_mod, C, reuse_a, reuse_b)
        acc = __builtin_amdgcn_wmma_f32_16x16x4_f32(
            false, a, false, b, (short)0, acc, false, false);
    }

    // D layout: lane L holds column N=L%16 (edge); VGPR r = row M = r + 8*half
    // (M = l). Lanes 0-15 own all 8 valid l outputs of their edge.
    if (half == 0 && e0 < E) {
        v4f lo = {acc[0], acc[1], acc[2], acc[3]};
        v4f hi = {acc[4], acc[5], acc[6], acc[7]};
        float* p = out + e0 * OUT_L;
        *(v4f*)(p)     = lo;                // global_store_b128
        *(v4f*)(p + 4) = hi;                // global_store_b128
    }
}

extern "C" void kernel_launch(void* const* d_in, const int* in_sizes, int n_in,
                              void* d_out, int out_size, void* d_ws, size_t ws_size,
                              hipStream_t stream) {
    const long long* edge_index = (const long long*)d_in[0];  // int64 [2, E]
    const float*     node_sh    = (const float*)d_in[1];      // f32 [N, 64]
    float*           out        = (float*)d_out;              // f32 [E, 8]

    const int E = in_sizes[0] / 2;
    const int tiles  = (E + 15) / 16;          // 16 edges per wave-tile
    const int blocks = (tiles + 7) / 8;        // 8 waves (256 thr) per block

    sh_irrep_inner_kernel<<<blocks, 256, 0, stream>>>(edge_index, node_sh, out, E);
}